// FocalLoss_12515534701332
// MI455X (gfx1250) — compile-verified
//
#include <hip/hip_runtime.h>

constexpr int   kB = 8, kA = 100000, kC = 80, kM = 32;
constexpr float kAlpha = 0.25f;
constexpr int   kT  = 256;
constexpr int   kWv = 32;                                // wave32
constexpr int   kWaves = kT / kWv;                       // 8 waves / block
constexpr int   kNB1 = (kA + kT - 1) / kT;               // 391 blocks (match)
constexpr int   kF4  = kA * kC / 4;                      // 2,000,000 float4 per image
constexpr int   kNTile  = 8;                             // pipelined tiles per block
constexpr int   kTileF4 = kT;                            // 256 float4 per tile
constexpr int   kBlkF4  = kTileF4 * kNTile;              // 2048 float4 per block
constexpr int   kNB2 = (kF4 + kBlkF4 - 1) / kBlkF4;      // 977 blocks (focal)

// ---------------- kernel 1: anchor<->GT matching + smooth-L1 ---------------
__launch_bounds__(256)
__global__ void fl_match(const float* __restrict__ regs,
                         const float* __restrict__ anchors,
                         const float* __restrict__ ann,
                         int* __restrict__ codes,
                         float* __restrict__ part_pos,
                         float* __restrict__ part_reg) {
  __shared__ float sann[kM * 5];
  __shared__ float sred[kT];
  const int b = blockIdx.y;
  const int a = blockIdx.x * kT + threadIdx.x;
  if (threadIdx.x < kM * 5) sann[threadIdx.x] = ann[b * kM * 5 + threadIdx.x];
  __syncthreads();

  float posf = 0.f, regl = 0.f;
  if (a < kA) {
    const float a0 = anchors[a * 4 + 0], a1 = anchors[a * 4 + 1];
    const float a2 = anchors[a * 4 + 2], a3 = anchors[a * 4 + 3];
    const float areaA = (a2 - a0) * (a3 - a1);
    float best = -2.0f; int bi = 0;            // every candidate >= -1
    for (int m = 0; m < kM; ++m) {
      const float g0 = sann[m * 5 + 0], g1 = sann[m * 5 + 1];
      const float g2 = sann[m * 5 + 2], g3 = sann[m * 5 + 3];
      const float lab = sann[m * 5 + 4];
      const float iw = fmaxf(fminf(a2, g2) - fmaxf(a0, g0), 0.f);
      const float ih = fmaxf(fminf(a3, g3) - fmaxf(a1, g1), 0.f);
      const float inter = iw * ih;
      const float ua = fmaxf(areaA + (g2 - g0) * (g3 - g1) - inter, 1e-8f);
      const float iou = (lab >= 0.f) ? (inter / ua) : -1.f;
      if (iou > best) { best = iou; bi = m; }  // strict > keeps first argmax
    }
    int code;
    if (best >= 0.5f) {                        // positive anchor
      code = (int)sann[bi * 5 + 4];
      posf = 1.f;
      const float aw = a2 - a0, ah = a3 - a1;
      const float acx = a0 + 0.5f * aw, acy = a1 + 0.5f * ah;
      const float g0 = sann[bi * 5 + 0], g1 = sann[bi * 5 + 1];
      const float g2 = sann[bi * 5 + 2], g3 = sann[bi * 5 + 3];
      const float gw0 = g2 - g0, gh0 = g3 - g1;
      const float gcx = g0 + 0.5f * gw0, gcy = g1 + 0.5f * gh0;
      const float gw = fmaxf(gw0, 1.f), gh = fmaxf(gh0, 1.f);
      float t[4];
      t[0] = ((gcx - acx) / aw) / 0.1f;
      t[1] = ((gcy - acy) / ah) / 0.1f;
      t[2] = logf(gw / aw) / 0.2f;
      t[3] = logf(gh / ah) / 0.2f;
      const float* r = regs + ((size_t)b * kA + a) * 4;
      const float4 rv = *reinterpret_cast<const float4*>(r);
      const float rr4[4] = {rv.x, rv.y, rv.z, rv.w};
#pragma unroll
      for (int j = 0; j < 4; ++j) {
        const float d = fabsf(t[j] - rr4[j]);
        regl += (d <= (1.f / 9.f)) ? (0.5f * 9.f * d * d) : (d - 0.5f / 9.f);
      }
    } else {
      code = (best < 0.4f) ? -1 : -2;          // -1 negative, -2 ignore
    }
    codes[(size_t)b * kA + a] = code;
  }

  const int t = threadIdx.x;
  sred[t] = posf; __syncthreads();
  for (int s = kT / 2; s > 0; s >>= 1) { if (t < s) sred[t] += sred[t + s]; __syncthreads(); }
  const float possum = sred[0];
  __syncthreads();
  sred[t] = regl; __syncthreads();
  for (int s = kT / 2; s > 0; s >>= 1) { if (t < s) sred[t] += sred[t + s]; __syncthreads(); }
  if (t == 0) {
    part_pos[b * kNB1 + blockIdx.x] = possum;
    part_reg[b * kNB1 + blockIdx.x] = sred[0];
  }
}

// ------- kernel 2: streaming focal loss, async global->LDS double buffer ---
__launch_bounds__(256)
__global__ void fl_focal(const float* __restrict__ cls,
                         const int* __restrict__ codes,
                         float* __restrict__ part_cls) {
  // per-wave double buffer: 8 waves * 2 bufs * 32 lanes * 16B = 8KB
  __shared__ __align__(16) float stage[kWaves * 2 * kWv * 4];
  __shared__ float sred[kT];
  const int b = blockIdx.y;
  const int t = threadIdx.x;
  const int wave = t >> 5;
  const int lane = t & (kWv - 1);
  const size_t imgBase = (size_t)b * (size_t)(kA * kC);
  const int f0 = blockIdx.x * kBlkF4 + t;        // f for tile 0; +kTileF4 per tile

  float* slot0 = &stage[((wave * 2 + 0) * kWv + lane) * 4];
  float* slot1 = &stage[((wave * 2 + 1) * kWv + lane) * 4];
  // ISA flat-aperture rule: LDS byte address = low 32 bits of generic pointer
  const unsigned lds0 = (unsigned)(uintptr_t)slot0;
  const unsigned lds1 = (unsigned)(uintptr_t)slot1;

  auto issue = [&](int tile, unsigned ldsAddr) {
    int f = f0 + tile * kTileF4;
    int fc = (f < kF4) ? f : (kF4 - 1);          // clamp tail: DMA stays in-bounds
    const float* g = cls + imgBase + (size_t)fc * 4;
    asm volatile("global_load_async_to_lds_b128 %0, %1, off"
                 :: "v"(ldsAddr), "v"(g) : "memory");
  };

  float sum = 0.f;
  issue(0, lds0);
  for (int tile = 0; tile < kNTile; ++tile) {
    float* curp = (tile & 1) ? slot1 : slot0;
    if (tile + 1 < kNTile) {
      issue(tile + 1, (tile & 1) ? lds0 : lds1); // prefetch next tile
      asm volatile("s_wait_asynccnt 0x1" ::: "memory");   // tile `tile` landed
    } else {
      asm volatile("s_wait_asynccnt 0x0" ::: "memory");
    }
    const int f = f0 + tile * kTileF4;
    if (f < kF4) {
      const int anchor = f / (kC / 4);
      const int code = codes[b * kA + anchor];
      if (code != -2) {                          // ignored anchors contribute 0
        const float4 v = *reinterpret_cast<const float4*>(curp);  // ds_load_b128
        const int c0 = (f % (kC / 4)) * 4;
        const float pv[4] = {v.x, v.y, v.z, v.w};
#pragma unroll
        for (int j = 0; j < 4; ++j) {
          const float p = fminf(fmaxf(pv[j], 1e-4f), 1.f - 1e-4f);
          const bool isPos = (c0 + j) == code;   // target==1 only at matched class
          const float w = isPos ? (kAlpha * (1.f - p) * (1.f - p)) : (0.75f * p * p);
          const float q = isPos ? p : (1.f - p); // one hw log per element
          sum += w * (-__logf(q));
        }
      }
    }
  }

  sred[t] = sum; __syncthreads();
  for (int s = kT / 2; s > 0; s >>= 1) { if (t < s) sred[t] += sred[t + s]; __syncthreads(); }
  if (t == 0) part_cls[b * kNB2 + blockIdx.x] = sred[0];
}

// ---------------- kernel 3: per-image deterministic reduction --------------
__launch_bounds__(256)
__global__ void fl_reduce(const float* __restrict__ part_cls,
                          const float* __restrict__ part_pos,
                          const float* __restrict__ part_reg,
                          const float* __restrict__ ann,
                          float* __restrict__ imgloss) {
  const int b = blockIdx.x;
  const int t = threadIdx.x;
  float c = 0.f, pp = 0.f, rr = 0.f;
  for (int i = t; i < kNB2; i += kT) c += part_cls[b * kNB2 + i];
  for (int i = t; i < kNB1; i += kT) { pp += part_pos[b * kNB1 + i]; rr += part_reg[b * kNB1 + i]; }
  __shared__ float sc[kT], sp[kT], sr[kT];
  sc[t] = c; sp[t] = pp; sr[t] = rr; __syncthreads();
  for (int s = kT / 2; s > 0; s >>= 1) {
    if (t < s) { sc[t] += sc[t + s]; sp[t] += sp[t + s]; sr[t] += sr[t + s]; }
    __syncthreads();
  }
  if (t == 0) {
    bool anyv = false;
    for (int m = 0; m < kM; ++m) anyv = anyv || (ann[b * kM * 5 + m * 5 + 4] >= 0.f);
    const float np = sp[0];
    const float clsl = sc[0] / fmaxf(np, 1.f);
    const float regl = (np > 0.f) ? (sr[0] / fmaxf(4.f * np, 1.f)) : 0.f;
    imgloss[b]      = anyv ? clsl : 0.f;
    imgloss[kB + b] = anyv ? regl : 0.f;
  }
}

// ---------------- kernel 4: batch mean -> d_out ----------------------------
__global__ void fl_final(const float* __restrict__ imgloss, float* __restrict__ out) {
  if (threadIdx.x == 0) {
    float c = 0.f, r = 0.f;
    for (int b = 0; b < kB; ++b) { c += imgloss[b]; r += imgloss[kB + b]; }
    out[0] = c / (float)kB;
    out[1] = r / (float)kB;
  }
}

extern "C" void kernel_launch(void* const* d_in, const int* in_sizes, int n_in,
                              void* d_out, int out_size, void* d_ws, size_t ws_size,
                              hipStream_t stream) {
  (void)in_sizes; (void)n_in; (void)out_size; (void)ws_size;
  const float* cls     = (const float*)d_in[0];   // [8,100000,80]
  const float* regs    = (const float*)d_in[1];   // [8,100000,4]
  const float* anchors = (const float*)d_in[2];   // [1,100000,4]
  const float* ann     = (const float*)d_in[3];   // [8,32,5]
  float* out = (float*)d_out;

  // workspace layout (all slots written every call; no init kernel needed)
  float* wsf      = (float*)d_ws;
  float* part_cls = wsf;                          // [kB * kNB2]
  float* part_pos = part_cls + kB * kNB2;         // [kB * kNB1]
  float* part_reg = part_pos + kB * kNB1;         // [kB * kNB1]
  float* imgloss  = part_reg + kB * kNB1;         // [2 * kB]
  int*   codes    = (int*)(imgloss + 2 * kB);     // [kB * kA]

  fl_match <<<dim3(kNB1, kB), kT, 0, stream>>>(regs, anchors, ann, codes, part_pos, part_reg);
  fl_focal <<<dim3(kNB2, kB), kT, 0, stream>>>(cls, codes, part_cls);
  fl_reduce<<<dim3(kB),       kT, 0, stream>>>(part_cls, part_pos, part_reg, ann, imgloss);
  fl_final <<<1, 32, 0, stream>>>(imgloss, out);
}